// _global_attention_sop_f_22814866277108
// MI455X (gfx1250) — compile-verified
//
#include <hip/hip_runtime.h>

// ---------------------------------------------------------------------------
// global_attention_sop: per-graph softmax-weighted Gram matrix, triu-vectorized
//   out[g] = triu( sum_{k in g} a_k * x_k x_k^T ),  a = scatter_softmax(x@w+b)
// Compute: V_WMMA_F32_16X16X4_F32 (fp32-exact, 4 nodes per WMMA), one
// workgroup (16 wave32) per graph, one 16x16 output tile per wave.
// Staging: GLOBAL_LOAD_ASYNC_TO_LDS_B128 via inline asm (ASYNCcnt tracked).
// ---------------------------------------------------------------------------

typedef __attribute__((ext_vector_type(2))) float v2f;
typedef __attribute__((ext_vector_type(8))) float v8f;

static constexpr int FD     = 64;                  // hidden dim
static constexpr int NTRIU  = FD * (FD + 1) / 2;   // 2080
static constexpr int CHUNK  = 64;                  // nodes staged per iteration
static constexpr int LDS_STRIDE = 72;              // 2*72 % 64 == 16 -> conflict-free

// Async global->LDS 128-bit copy. VDST holds the LDS byte address (low 32 bits
// of the generic pointer: flat addresses truncate to the LDS offset), VADDR is
// the 64-bit global address. Tracked by ASYNCcnt.
__device__ __forceinline__ void stage16(float* dst_lds, const float* src_glb) {
  unsigned lds_off = (unsigned)(size_t)dst_lds;
  unsigned long long gaddr = (unsigned long long)(size_t)src_glb;
  asm volatile("global_load_async_to_lds_b128 %0, %1, off"
               :
               : "v"(lds_off), "v"(gaddr)
               : "memory");
}

__device__ __forceinline__ void stage_wait() {
#if defined(__has_builtin) && __has_builtin(__builtin_amdgcn_s_wait_asynccnt)
  __builtin_amdgcn_s_wait_asynccnt(0);
#else
  asm volatile("s_wait_asynccnt 0" ::: "memory");
#endif
}

// order-preserving float<->uint mapping for atomic max
__device__ __forceinline__ unsigned fmap(float f) {
  unsigned u = __float_as_uint(f);
  return (u & 0x80000000u) ? ~u : (u | 0x80000000u);
}
__device__ __forceinline__ float funmap(unsigned u) {
  u = (u & 0x80000000u) ? (u & 0x7FFFFFFFu) : ~u;
  return __uint_as_float(u);
}

__device__ __forceinline__ int lower_bound(const int* __restrict__ a, int n, int v) {
  int lo = 0, hi = n;
  while (lo < hi) { int mid = (lo + hi) >> 1; if (a[mid] < v) lo = mid + 1; else hi = mid; }
  return lo;
}

// ---- phase 1: imp[k] = dot(x[k], w) + b ------------------------------------
__global__ void k_imp(const float* __restrict__ x, const float* __restrict__ w,
                      const float* __restrict__ b, float* __restrict__ imp, int M) {
  int k = blockIdx.x * blockDim.x + threadIdx.x;
  if (k >= M) return;
  const float4* xr = (const float4*)(x + (size_t)k * FD);
  const float4* wr = (const float4*)w;
  float s = 0.f;
#pragma unroll
  for (int i = 0; i < FD / 4; ++i) {
    float4 a = xr[i], c = wr[i];
    s += a.x * c.x + a.y * c.y + a.z * c.z + a.w * c.w;
  }
  imp[k] = s + b[0];
}

// ---- phase 2: init per-graph reductions ------------------------------------
__global__ void k_init(unsigned* __restrict__ segmax_u, float* __restrict__ denom, int G) {
  int g = blockIdx.x * blockDim.x + threadIdx.x;
  if (g >= G) return;
  segmax_u[g] = 0u;       // < fmap(-inf): valid identity for max
  denom[g]    = 0.f;
}

// ---- phase 3: segment max ---------------------------------------------------
__global__ void k_segmax(const float* __restrict__ imp, const int* __restrict__ batch,
                         unsigned* __restrict__ segmax_u, int M) {
  int k = blockIdx.x * blockDim.x + threadIdx.x;
  if (k >= M) return;
  atomicMax(&segmax_u[batch[k]], fmap(imp[k]));
}

// ---- phase 4: e = exp(imp - max); denom += e (in-place over imp) -----------
__global__ void k_exp(float* __restrict__ val, const int* __restrict__ batch,
                      const unsigned* __restrict__ segmax_u, float* __restrict__ denom, int M) {
  int k = blockIdx.x * blockDim.x + threadIdx.x;
  if (k >= M) return;
  int g = batch[k];
  float e = __expf(val[k] - funmap(segmax_u[g]));
  val[k] = e;
  atomicAdd(&denom[g], e);
}

// ---- phase 5: per-graph WMMA Gram + triu scatter ---------------------------
// grid = G blocks of 512 threads (16 wave32). Wave w -> output tile (w/4, w%4).
__launch_bounds__(512)
__global__ void k_gram(const float* __restrict__ x, const float* __restrict__ e,
                       const float* __restrict__ denom, const int* __restrict__ batch,
                       float* __restrict__ out, int M) {
  const int g = blockIdx.x;

  __shared__ int   sb[2];
  __shared__ float lds_a[CHUNK];                 // softmax weights a_k
  __shared__ float lds_x[CHUNK * LDS_STRIDE];    // x rows
  __shared__ float lds_xa[CHUNK * LDS_STRIDE];   // a_k * x rows (B operand)

  if (threadIdx.x == 0) sb[0] = lower_bound(batch, M, g);
  if (threadIdx.x == 1) sb[1] = lower_bound(batch, M, g + 1);
  __syncthreads();
  const int s0 = sb[0], s1 = sb[1];
  const float inv_denom = 1.0f / denom[g];       // unused if segment empty

  const int wave = threadIdx.x >> 5;             // 0..15
  const int lane = threadIdx.x & 31;
  const int row0 = (wave >> 2) * 16;             // output tile row origin
  const int col0 = (wave & 3) * 16;              // output tile col origin
  const int m    = lane & 15;                    // fragment index within lane-half
  const int kh   = lane >> 4;                    // lane half -> K pair {2kh, 2kh+1}

  v8f acc = {};                                  // 16x16 f32 accumulator tile

  for (int base = s0; base < s1; base += CHUNK) {
    const int cn  = min(CHUNK, s1 - base);
    const int cnp = (cn + 7) & ~7;               // pad to multiple of 8 nodes
    __syncthreads();                             // previous-iter consumers done

    // ---- stage x slab (contiguous: batch is sorted) via async B128 copies
    const float* src = x + (size_t)base * FD;
    for (int idx = threadIdx.x; idx < cn * (FD / 4); idx += 512) {
      const int node = idx >> 4, part = idx & 15;        // 16 float4s per row
      stage16(&lds_x[node * LDS_STRIDE + part * 4], src + idx * 4);
    }
    // zero the padded rows so padded K-slices contribute exactly 0
    for (int idx = cn * FD + threadIdx.x; idx < cnp * FD; idx += 512) {
      lds_x[(idx >> 6) * LDS_STRIDE + (idx & 63)] = 0.f;
    }
    for (int idx = threadIdx.x; idx < cnp; idx += 512) {
      lds_a[idx] = (idx < cn) ? e[base + idx] * inv_denom : 0.f;
    }
    stage_wait();                                // drain this wave's ASYNCcnt
    __syncthreads();

    // ---- pre-scale B operand: xa = a_k * x_k
    for (int idx = threadIdx.x; idx < cnp * FD; idx += 512) {
      const int node = idx >> 6, d = idx & 63;
      lds_xa[node * LDS_STRIDE + d] = lds_a[node] * lds_x[node * LDS_STRIDE + d];
    }
    __syncthreads();

    // ---- K-loop: 8 nodes (2 WMMAs) per iteration, no divergence
    const float* xr = &lds_x [(2 * kh) * LDS_STRIDE + row0 + m];
    const float* xc = &lds_xa[(2 * kh) * LDS_STRIDE + col0 + m];
    for (int k0 = 0; k0 < cnp; k0 += 8) {
      const float* r = xr + k0 * LDS_STRIDE;
      const float* c = xc + k0 * LDS_STRIDE;
      v2f A, B;
      A.x = r[0];               A.y = r[LDS_STRIDE];
      B.x = c[0];               B.y = c[LDS_STRIDE];
      acc = __builtin_amdgcn_wmma_f32_16x16x4_f32(false, A, false, B,
                                                  (short)0, acc, false, false);
      A.x = r[4 * LDS_STRIDE];  A.y = r[5 * LDS_STRIDE];
      B.x = c[4 * LDS_STRIDE];  B.y = c[5 * LDS_STRIDE];
      acc = __builtin_amdgcn_wmma_f32_16x16x4_f32(false, A, false, B,
                                                  (short)0, acc, false, false);
    }
  }

  // C/D layout: VGPR r -> M = r + 8*kh, N = m. Scatter upper triangle only.
  const int nj = col0 + m;
  float* og = out + (size_t)g * NTRIU;
#pragma unroll
  for (int r = 0; r < 8; ++r) {
    const int mi = row0 + r + 8 * kh;
    if (nj >= mi) {
      const int off = mi * FD - (mi * (mi - 1)) / 2 + (nj - mi);
      og[off] = acc[r];
    }
  }
}

// ---------------------------------------------------------------------------
extern "C" void kernel_launch(void* const* d_in, const int* in_sizes, int n_in,
                              void* d_out, int out_size, void* d_ws, size_t ws_size,
                              hipStream_t stream) {
  const float* x     = (const float*)d_in[0];   // [M, 64]
  const float* att_w = (const float*)d_in[1];   // [1, 64]
  const float* att_b = (const float*)d_in[2];   // [1]
  const int*   batch = (const int*)d_in[3];     // [M] sorted graph ids
  // d_in[4] = edge: dead input in the reference.
  float* out = (float*)d_out;

  const int M = in_sizes[0] / FD;
  const int G = out_size / NTRIU;

  // workspace: val[M] (imp -> e), segmax_u[G], denom[G]
  float*    val      = (float*)d_ws;
  unsigned* segmax_u = (unsigned*)((char*)d_ws + (size_t)M * sizeof(float));
  float*    denom    = (float*)((char*)d_ws + (size_t)M * sizeof(float)
                                            + (size_t)G * sizeof(unsigned));

  if (M <= 0 || G <= 0) return;

  k_imp   <<<(M + 255) / 256, 256, 0, stream>>>(x, att_w, att_b, val, M);
  k_init  <<<(G + 255) / 256, 256, 0, stream>>>(segmax_u, denom, G);
  k_segmax<<<(M + 255) / 256, 256, 0, stream>>>(val, batch, segmax_u, M);
  k_exp   <<<(M + 255) / 256, 256, 0, stream>>>(val, batch, segmax_u, denom, M);
  k_gram  <<<G, 512, 0, stream>>>(x, val, denom, batch, out, M);
}